// memory_gate_4483945857627
// MI455X (gfx1250) — compile-verified
//
#include <hip/hip_runtime.h>
#include <hip/hip_bf16.h>
#include <math.h>

// ---------------------------------------------------------------------------
// MI455X plan:
//  * scores = embed_memory @ (Kw @ Q) -- the only big op; pure HBM bandwidth
//    (512MB read -> ~22us @ 23.3TB/s). Done with v_wmma_f32_16x16x4_f32,
//    16 rows/wave, kq broadcast into all 16 B columns (FLOPs are free).
//  * full V GEMM is NOT computed: only the 5 top-k rows are needed.
//  * softmax stats via ordered-uint atomicMax + atomicAdd sum-of-exp.
//  * top-5 via 5 grid-wide u64 atomicMax selection passes (ties -> low idx).
//  * tiny single-block tail: gate MLP, layernorms, actor MLP.
// ---------------------------------------------------------------------------

#define H 128
#define M_TOTAL 1000000
#define K_ATTN 5
#define A_OUT 7
#define GH 64

typedef __attribute__((ext_vector_type(2))) float v2f;
typedef __attribute__((ext_vector_type(8))) float v8f;

__device__ __forceinline__ unsigned ordered_key(float f) {
  unsigned b = __float_as_uint(f);
  return (b & 0x80000000u) ? ~b : (b | 0x80000000u);
}
__device__ __forceinline__ float from_ordered(unsigned u) {
  unsigned b = (u & 0x80000000u) ? (u ^ 0x80000000u) : ~u;
  return __uint_as_float(b);
}

// ws layout (floats): [0..127] kq_scaled, [128] gmax(ordered u32),
// [129] sumexp, [132..141] 5 x u64 topk keys (byte offset 528, 8B aligned).

// ---------------------------------------------------------------- kernel 1
__global__ void __launch_bounds__(128) mg_setup(
    const float* __restrict__ qs, const float* __restrict__ Qw,
    const float* __restrict__ Kw, float* __restrict__ wsf,
    unsigned* __restrict__ wsu, unsigned long long* __restrict__ topk) {
  __shared__ float Qs[H];
  int t = threadIdx.x;  // 128 threads
  float acc = 0.f;
  for (int h = 0; h < H; ++h) acc += qs[h] * Qw[h * H + t];
  Qs[t] = acc;
  __syncthreads();
  float kq = 0.f;
  for (int c = 0; c < H; ++c) kq += Kw[t * H + c] * Qs[c];
  const float invscale = 0.04419417382415922f;  // 1/(TAU*sqrt(H)) = 1/(2*sqrt(128))
  wsf[t] = kq * invscale;
  if (t == 0) { wsu[128] = 0u; wsf[129] = 0.f; }
  if (t < K_ATTN) topk[t] = 0ull;
}

// ---------------------------------------------------------------- kernel 2
// 256 threads = 8 waves; each wave computes 16 rows of scores with WMMA.
__global__ void __launch_bounds__(256) mg_scores(
    const float* __restrict__ e, const float* __restrict__ noise,
    const float* __restrict__ kq_g, unsigned* __restrict__ gmax,
    float* __restrict__ scores_out) {
  __shared__ float kq[H];
  __shared__ unsigned wmax[8];
  int tid = threadIdx.x;
  if (tid < H) kq[tid] = kq_g[tid];
  __syncthreads();

  int wave = tid >> 5;
  int lane = tid & 31;
  long long rows_base = (long long)blockIdx.x * 128 + (long long)wave * 16;
  unsigned my_key = 0u;

  if (rows_base < M_TOTAL) {  // M % 16 == 0 -> waves are all-valid or all-invalid
    int half = lane >> 4;     // 0: K = k,k+1   1: K = k+2,k+3
    int mrow = lane & 15;
    const float* erow = e + (rows_base + mrow) * H + 2 * half;
    v8f c = {};
#pragma unroll
    for (int k = 0; k < H; k += 4) {
      v2f a, b;
      a.x = erow[k];
      a.y = erow[k + 1];
      b.x = kq[k + 2 * half];
      b.y = kq[k + 2 * half + 1];
      // D[m,n] += sum_k A[m,k]*B[k,n]; B columns identical => D[m,*] = s_m
      c = __builtin_amdgcn_wmma_f32_16x16x4_f32(false, a, false, b, (short)0, c,
                                                false, false);
    }
    // lane N==0 of each half owns rows base+8*half .. base+8*half+7 in c[0..7]
    if (mrow == 0) {
      long long r0 = rows_base + 8 * half;
#pragma unroll
      for (int j = 0; j < 8; ++j) {
        float s = c[j];
        scores_out[r0 + j] = s;
        float tv = s + 0.001f * noise[r0 + j];
        unsigned k2 = ordered_key(tv);
        if (k2 > my_key) my_key = k2;
      }
    }
  }
#pragma unroll
  for (int off = 16; off >= 1; off >>= 1) {
    unsigned o = __shfl_xor(my_key, off, 32);
    if (o > my_key) my_key = o;
  }
  if (lane == 0) wmax[wave] = my_key;
  __syncthreads();
  if (tid == 0) {
    unsigned m = wmax[0];
    for (int w = 1; w < 8; ++w) m = (wmax[w] > m) ? wmax[w] : m;
    atomicMax(gmax, m);
  }
}

// ---------------------------------------------------------------- kernel 3
__global__ void __launch_bounds__(256) mg_sumexp(
    const float* __restrict__ scores, const float* __restrict__ noise,
    const unsigned* __restrict__ gmaxp, float* __restrict__ sum_out) {
  __shared__ float red[256];
  float gmax = from_ordered(*gmaxp);
  float acc = 0.f;
  for (int i = blockIdx.x * blockDim.x + threadIdx.x; i < M_TOTAL;
       i += gridDim.x * blockDim.x) {
    float tv = scores[i] + 0.001f * noise[i];
    acc += __expf(tv - gmax);
  }
  red[threadIdx.x] = acc;
  __syncthreads();
  for (int s = 128; s >= 1; s >>= 1) {
    if (threadIdx.x < s) red[threadIdx.x] += red[threadIdx.x + s];
    __syncthreads();
  }
  if (threadIdx.x == 0) atomicAdd(sum_out, red[0]);
}

// ---------------------------------------------------------------- kernel 4
// round r: grid-wide argmax excluding already-chosen indices.
__global__ void __launch_bounds__(256) mg_select(
    const float* __restrict__ scores, const float* __restrict__ noise,
    unsigned long long* __restrict__ topk, int round) {
  __shared__ unsigned long long red[256];
  unsigned chosen[K_ATTN];
  for (int r = 0; r < round; ++r)
    chosen[r] = 0xFFFFFFFFu - (unsigned)(topk[r] & 0xFFFFFFFFull);
  unsigned long long best = 0ull;
  for (int i = blockIdx.x * blockDim.x + threadIdx.x; i < M_TOTAL;
       i += gridDim.x * blockDim.x) {
    bool skip = false;
    for (int r = 0; r < round; ++r) skip |= (chosen[r] == (unsigned)i);
    if (skip) continue;
    float tv = scores[i] + 0.001f * noise[i];
    unsigned long long key = ((unsigned long long)ordered_key(tv) << 32) |
                             (unsigned long long)(0xFFFFFFFFu - (unsigned)i);
    if (key > best) best = key;
  }
  red[threadIdx.x] = best;
  __syncthreads();
  for (int s = 128; s >= 1; s >>= 1) {
    if (threadIdx.x < s && red[threadIdx.x + s] > red[threadIdx.x])
      red[threadIdx.x] = red[threadIdx.x + s];
    __syncthreads();
  }
  if (threadIdx.x == 0) atomicMax(topk + round, red[0]);
}

// ---------------------------------------------------------------- kernel 5
__global__ void __launch_bounds__(256) mg_final(
    const float* __restrict__ hidden_state, const float* __restrict__ hm,
    const float* __restrict__ am, const float* __restrict__ rm,
    const int* __restrict__ mi, const float* __restrict__ Vw,
    const float* __restrict__ gw1, const float* __restrict__ gb1,
    const float* __restrict__ gw2, const float* __restrict__ gb2,
    const float* __restrict__ lng, const float* __restrict__ lnb,
    const float* __restrict__ aw1, const float* __restrict__ ab1,
    const float* __restrict__ aw2, const float* __restrict__ ab2,
    const unsigned* __restrict__ gmaxp, const float* __restrict__ sumexp_p,
    const unsigned long long* __restrict__ topk, float* __restrict__ out) {
  __shared__ float Wtop[K_ATTN];
  __shared__ int idx[K_ATTN];
  __shared__ float hbar[H];
  __shared__ float attn[H];
  __shared__ float red[256];
  __shared__ float cv[4 * H];
  __shared__ float pre1[GH];
  __shared__ float a1[GH];
  __shared__ float feat[H + 2 * K_ATTN];
  __shared__ float stat[4];
  __shared__ float alpha_s;
  int tid = threadIdx.x;

  if (tid < K_ATTN) {
    unsigned long long key = topk[tid];
    int i = (int)(0xFFFFFFFFu - (unsigned)(key & 0xFFFFFFFFull));
    idx[tid] = i;
    float tv = from_ordered((unsigned)(key >> 32));
    Wtop[tid] = __expf(tv - from_ordered(*gmaxp)) / (*sumexp_p);
  }
  __syncthreads();

  // attn = (Wtop @ H_top) @ Vw  -- only 5 rows of the V projection needed
  if (tid < H) {
    float a = 0.f;
    for (int k = 0; k < K_ATTN; ++k)
      a += Wtop[k] * hm[(long long)idx[k] * H + tid];
    hbar[tid] = a;
  }
  __syncthreads();
  if (tid < H) {
    float a = 0.f;
    for (int h = 0; h < H; ++h) a += hbar[h] * Vw[h * H + tid];
    attn[tid] = a;
  }
  __syncthreads();

  float hv = (tid < H) ? hidden_state[tid] : 0.f;
  float av = (tid < H) ? attn[tid] : 0.f;

  // layernorm stats (mean/var over 128) for hidden_state and attn
  red[tid] = hv; __syncthreads();
  for (int s = 128; s >= 1; s >>= 1) { if (tid < s) red[tid] += red[tid + s]; __syncthreads(); }
  if (tid == 0) stat[0] = red[0] / (float)H;
  __syncthreads();
  float dh = (tid < H) ? hv - stat[0] : 0.f;
  red[tid] = dh * dh; __syncthreads();
  for (int s = 128; s >= 1; s >>= 1) { if (tid < s) red[tid] += red[tid + s]; __syncthreads(); }
  if (tid == 0) stat[1] = red[0] / (float)H;
  __syncthreads();
  red[tid] = av; __syncthreads();
  for (int s = 128; s >= 1; s >>= 1) { if (tid < s) red[tid] += red[tid + s]; __syncthreads(); }
  if (tid == 0) stat[2] = red[0] / (float)H;
  __syncthreads();
  float da = (tid < H) ? av - stat[2] : 0.f;
  red[tid] = da * da; __syncthreads();
  for (int s = 128; s >= 1; s >>= 1) { if (tid < s) red[tid] += red[tid + s]; __syncthreads(); }
  if (tid == 0) stat[3] = red[0] / (float)H;
  __syncthreads();

  // gate MLP on concat [h, attn, h-attn, h*attn]
  if (tid < H) {
    cv[tid] = hv; cv[H + tid] = av;
    cv[2 * H + tid] = hv - av; cv[3 * H + tid] = hv * av;
  }
  __syncthreads();
  if (tid < GH) {
    float a = gb1[tid];
    for (int j = 0; j < 4 * H; ++j) a += cv[j] * gw1[j * GH + tid];
    pre1[tid] = fmaxf(a, 0.f);
  }
  __syncthreads();
  if (tid == 0) {
    float a = gb2[0];
    for (int g = 0; g < GH; ++g) a += pre1[g] * gw2[g];
    alpha_s = 1.f / (1.f + __expf(-a));
  }
  __syncthreads();
  float alpha = alpha_s;

  if (tid < H) {
    float lnh = dh * rsqrtf(stat[1] + 1e-5f) * lng[tid] + lnb[tid];
    float lna = da * rsqrtf(stat[3] + 1e-5f) * lng[tid] + lnb[tid];
    feat[tid] = (1.f - alpha) * lnh + alpha * lna;
  }
  if (tid < K_ATTN) {
    feat[H + tid] = am[idx[tid]];
    feat[H + K_ATTN + tid] = rm[idx[tid]];
  }
  __syncthreads();

  // actor MLP: relu(relu(feat) @ aw1 + ab1) @ aw2 + ab2
  if (tid < GH) {
    float a = ab1[tid];
    for (int j = 0; j < H + 2 * K_ATTN; ++j)
      a += fmaxf(feat[j], 0.f) * aw1[j * GH + tid];
    a1[tid] = fmaxf(a, 0.f);
  }
  __syncthreads();
  if (tid < A_OUT) {
    float a = ab2[tid];
    for (int g = 0; g < GH; ++g) a += a1[g] * aw2[g * A_OUT + tid];
    out[tid] = a;  // logits
  }
  if (tid < K_ATTN) {
    out[7 + tid] = Wtop[tid];                  // W_top
    out[12 + tid] = (float)mi[idx[tid]];       // chosen_ids (as float)
  }
  if (tid == 0) out[17] = alpha;               // alpha
  // scores written directly by mg_scores at out+18
}

// ---------------------------------------------------------------------------
extern "C" void kernel_launch(void* const* d_in, const int* in_sizes, int n_in,
                              void* d_out, int out_size, void* d_ws,
                              size_t ws_size, hipStream_t stream) {
  const float* qs  = (const float*)d_in[0];
  const float* hs  = (const float*)d_in[1];
  const float* em  = (const float*)d_in[2];
  const float* hm  = (const float*)d_in[3];
  const float* am  = (const float*)d_in[4];
  const float* rm  = (const float*)d_in[5];
  const int*   mi  = (const int*)d_in[6];
  const float* nz  = (const float*)d_in[7];
  const float* Qw  = (const float*)d_in[8];
  const float* Kw  = (const float*)d_in[9];
  const float* Vw  = (const float*)d_in[10];
  const float* gw1 = (const float*)d_in[11];
  const float* gb1 = (const float*)d_in[12];
  const float* gw2 = (const float*)d_in[13];
  const float* gb2 = (const float*)d_in[14];
  const float* lng = (const float*)d_in[15];
  const float* lnb = (const float*)d_in[16];
  const float* aw1 = (const float*)d_in[17];
  const float* ab1 = (const float*)d_in[18];
  const float* aw2 = (const float*)d_in[19];
  const float* ab2 = (const float*)d_in[20];

  float* out = (float*)d_out;
  float* wsf = (float*)d_ws;
  unsigned* wsu = (unsigned*)d_ws;
  unsigned long long* topk = (unsigned long long*)(wsf + 132);
  float* scores_out = out + 18;

  mg_setup<<<1, 128, 0, stream>>>(qs, Qw, Kw, wsf, wsu, topk);
  mg_scores<<<(M_TOTAL + 127) / 128, 256, 0, stream>>>(em, nz, wsf, wsu + 128,
                                                       scores_out);
  mg_sumexp<<<1024, 256, 0, stream>>>(scores_out, nz, wsu + 128, wsf + 129);
  for (int r = 0; r < K_ATTN; ++r)
    mg_select<<<512, 256, 0, stream>>>(scores_out, nz, topk, r);
  mg_final<<<1, 256, 0, stream>>>(hs, hm, am, rm, mi, Vw, gw1, gb1, gw2, gb2,
                                  lng, lnb, aw1, ab1, aw2, ab2, wsu + 128,
                                  wsf + 129, topk, out);
}